// AnchorLoss_62388694942452
// MI455X (gfx1250) — compile-verified
//
#include <hip/hip_runtime.h>
#include <hip/hip_bf16.h>
#include <math.h>

// Problem constants (from reference): anchors [8192, 8, 512] f32
// x = reshape -> [NROWS=8192, NCOLS=4096]
#define NROWS 8192
#define NCOLS 4096

#define COLS_PER_BLOCK 128   // 8 waves x 16 cols
#define ROWS_PER_BLOCK 256
#define CHUNK_ROWS 32
#define NCHUNKS (ROWS_PER_BLOCK / CHUNK_ROWS)   // 8 (even -> unroll-by-2 works)
#define ROWSTRIDE 144        // dwords; 144 % 64 == 16 -> adjacent rows hit disjoint LDS banks

typedef float v2f __attribute__((ext_vector_type(2)));
typedef float v8f __attribute__((ext_vector_type(8)));

// ---------------------------------------------------------------------------
// CDNA5 async global->LDS copy (ASYNCcnt-tracked DMA, GVS addressing):
//   copies 16 bytes per lane from the global address sbase plus voff to the LDS offset in lds_off
// ---------------------------------------------------------------------------
__device__ __forceinline__ void async_load_b128(unsigned lds_off, unsigned voff,
                                                unsigned long long sbase) {
    asm volatile("global_load_async_to_lds_b128 %0, %1, %2 offset:0"
                 :: "v"(lds_off), "v"(voff), "s"(sbase)
                 : "memory");
}
__device__ __forceinline__ void wait_async_le(void) {   // current chunk done, next in flight
    asm volatile("s_wait_asynccnt 0x4" ::: "memory");
}
__device__ __forceinline__ void wait_async_all(void) {  // drain
    asm volatile("s_wait_asynccnt 0x0" ::: "memory");
}

// ---------------------------------------------------------------------------
// Kernel 0: zero the workspace accumulators (colsum[NCOLS] + sumsq scalar)
// ---------------------------------------------------------------------------
__global__ void anchor_zero_ws(float* __restrict__ ws) {
    int i = blockIdx.x * blockDim.x + threadIdx.x;
    if (i < NCOLS + 1) ws[i] = 0.0f;
}

// ---------------------------------------------------------------------------
// Kernel 1: streaming pass, double-buffered async DMA + WMMA.
//   - GLOBAL_LOAD_ASYNC_TO_LDS_B128 overlaps chunk c+1 DMA with chunk c WMMA
//   - colsum via V_WMMA_F32_16X16X4_F32 with A = ones(16x4):
//       D[m][n] = sum_k B[k][n]  (layout-robust: invariant to K mapping)
//   - sumsq via VALU FMA on the same staged values
// ---------------------------------------------------------------------------
__global__ void __launch_bounds__(256)
anchor_main(const float* __restrict__ x,
            float* __restrict__ colsum,   // ws[0..NCOLS)
            float* __restrict__ sumsq) {  // ws[NCOLS]
    __shared__ float tile[2][CHUNK_ROWS * ROWSTRIDE];

    const int tid  = threadIdx.x;
    const int lane = tid & 31;
    const int wave = tid >> 5;           // 0..7
    const int colbase = blockIdx.x * COLS_PER_BLOCK;
    const int rowbase = blockIdx.y * ROWS_PER_BLOCK;

    // staging coordinates: each wave DMAs one 128-col row as 32 x b128
    const int s_col = lane;              // float4 index 0..31 within the 128-col tile
    const int s_row = wave;              // base row 0..7, stepped by 8

    // compute coordinates: wave owns 16 columns; half-waves cover row pairs
    const int c = wave * 16 + (lane & 15);   // column within tile
    const int h = lane >> 4;                 // 0 for lanes 0-15, 1 for lanes 16-31

    // LDS byte offsets of the two buffers (generic shared ptr low 32 bits)
    const unsigned tb0 = (unsigned)(uintptr_t)&tile[0][0];
    const unsigned tb1 = (unsigned)(uintptr_t)&tile[1][0];
    const unsigned long long gbase = (unsigned long long)(uintptr_t)x;

    const v2f onesA = {1.0f, 1.0f};          // A = 16x4 all-ones (layout-invariant)
    v8f acc = {};                            // colsum accumulator (all rows of D equal)
    float ssq = 0.0f;

    // issue the 4 async b128 loads of one 32x128 chunk into buffer base `tb`
    auto issue = [&](int chunk, unsigned tb) {
        const int r0 = rowbase + chunk * CHUNK_ROWS;
        #pragma unroll
        for (int r4 = 0; r4 < 4; ++r4) {
            const int row = r4 * 8 + s_row;  // 0..31
            const unsigned goff = (unsigned)(((unsigned)(r0 + row) * NCOLS
                                              + colbase + s_col * 4) * sizeof(float));
            const unsigned loff = tb + (unsigned)((row * ROWSTRIDE + s_col * 4)
                                                  * sizeof(float));
            async_load_b128(loff, goff, gbase);
        }
    };

    // WMMA + sumsq over one staged 32x128 tile
    auto compute = [&](const float* t) {
        #pragma unroll
        for (int kb = 0; kb < CHUNK_ROWS / 4; ++kb) {
            v2f b;
            // rows kb*4 + {0,1} via component x (halves), kb*4 + {2,3} via y
            b.x = t[(kb * 4 + 0 + h) * ROWSTRIDE + c];
            b.y = t[(kb * 4 + 2 + h) * ROWSTRIDE + c];
            ssq += b.x * b.x + b.y * b.y;    // every element touched exactly once
            // (neg_a, A, neg_b, B, c_mod, C, reuse_a, reuse_b)
            acc = __builtin_amdgcn_wmma_f32_16x16x4_f32(
                      false, onesA, false, b, (short)0, acc, false, false);
        }
    };

    issue(0, tb0);                           // prefetch chunk 0
    for (int chunk = 0; chunk < NCHUNKS; chunk += 2) {
        // ---- phase A: compute buffer 0, DMA into buffer 1 ----
        if (chunk + 1 < NCHUNKS) { issue(chunk + 1, tb1); wait_async_le(); }
        else                     { wait_async_all(); }
        __syncthreads();
        compute(&tile[0][0]);
        __syncthreads();
        // ---- phase B: compute buffer 1, DMA into buffer 0 ----
        if (chunk + 2 < NCHUNKS) { issue(chunk + 2, tb0); wait_async_le(); }
        else                     { wait_async_all(); }
        __syncthreads();
        compute(&tile[1][0]);
        __syncthreads();
    }

    // acc[0] in lanes 0..15 holds colsum over this block's rows for column
    // colbase + wave*16 + lane  (lanes 16..31 hold duplicates -> skip)
    if (lane < 16) {
        atomicAdd(&colsum[colbase + wave * 16 + lane], acc[0]);
    }

    // wave32 reduction of partial sum-of-squares, one atomic per wave
    #pragma unroll
    for (int off = 16; off > 0; off >>= 1)
        ssq += __shfl_xor(ssq, off, 32);
    if (lane == 0) atomicAdd(sumsq, ssq);
}

// ---------------------------------------------------------------------------
// Kernel 2: finalize  loss = -(2*N*sumsq - 2*sum(colsum^2)) / sqrt(512)
// ---------------------------------------------------------------------------
__global__ void __launch_bounds__(256)
anchor_final(const float* __restrict__ ws, float* __restrict__ out) {
    const int tid = threadIdx.x;
    float csq = 0.0f;
    for (int i = tid; i < NCOLS; i += 256) {
        const float v = ws[i];
        csq += v * v;
    }
    #pragma unroll
    for (int off = 16; off > 0; off >>= 1)
        csq += __shfl_xor(csq, off, 32);

    __shared__ float part[8];
    if ((tid & 31) == 0) part[tid >> 5] = csq;
    __syncthreads();

    if (tid == 0) {
        float tot = 0.0f;
        #pragma unroll
        for (int w = 0; w < 8; ++w) tot += part[w];
        const float ssq   = ws[NCOLS];
        const float total = 2.0f * (float)NROWS * ssq - 2.0f * tot;
        out[0] = -total / sqrtf(512.0f);   // factor = sqrt(dim_emb)
    }
}

// ---------------------------------------------------------------------------
extern "C" void kernel_launch(void* const* d_in, const int* in_sizes, int n_in,
                              void* d_out, int out_size, void* d_ws, size_t ws_size,
                              hipStream_t stream) {
    const float* x = (const float*)d_in[0];      // anchors, 8192*8*512 f32
    float* out = (float*)d_out;                  // 1 f32
    float* ws  = (float*)d_ws;                   // ws[0..4095]=colsum, ws[4096]=sumsq
    (void)in_sizes; (void)n_in; (void)out_size; (void)ws_size;

    anchor_zero_ws<<<(NCOLS + 1 + 255) / 256, 256, 0, stream>>>(ws);

    dim3 grid(NCOLS / COLS_PER_BLOCK, NROWS / ROWS_PER_BLOCK);  // 32 x 32
    anchor_main<<<grid, 256, 0, stream>>>(x, ws, ws + NCOLS);

    anchor_final<<<1, 256, 0, stream>>>(ws, out);
}